// OHEMloss_5050881540318
// MI455X (gfx1250) — compile-verified
//
#include <hip/hip_runtime.h>
#include <stdint.h>

// OHEM loss for MI455X (gfx1250).
// Memory-bound selection/reduction: 75.5 MB input, L2-resident (192 MB L2).
// Top-k-sum via exact 3-level radix select on float bit patterns using LDS
// (count,sum) histograms. Common case: threshold-bin elements are compacted
// into LDS after pass 1, so only TWO global passes are needed.
// CDNA5 paths: GLOBAL_LOAD_ASYNC_TO_LDS_B128 double-buffered staging
// (ASYNCcnt + s_wait_asynccnt), ds_add_u32/ds_add_f32 LDS atomics,
// wave32 shuffles, global_prefetch_b8.

#define N_PIX      (384 * 384)    // 147456 pixels per batch row
#define NBINS      2048
#define NTHREADS   512
#define NWAVES     (NTHREADS / 32)
#define RATIO_U    3u
#define TOPK_U     50u
#define STANDARD_F 0.1f
#define CAP        8192u          // LDS compaction capacity (floats)
#define ARENA_B    32768          // bytes: staging (pass1) / cbuf (pass2+)

__device__ __forceinline__ void zero_hist(unsigned* s_cnt, float* s_sum) {
  for (int i = threadIdx.x; i < NBINS; i += NTHREADS) {
    s_cnt[i] = 0u;
    s_sum[i] = 0.0f;
  }
}

// Thread-0 scan from top bin down. Returns threshold bin, remaining k inside
// it, and the running sum of everything strictly above it.
__device__ __forceinline__ void scan_level(const unsigned* s_cnt, const float* s_sum,
                                           int nbins, unsigned krem_in, double sgt_in,
                                           unsigned* bsel_out, unsigned* krem_out,
                                           double* sgt_out) {
  unsigned c = 0u;
  int bsel = 0;
  double sgt = sgt_in;
  for (int bb = nbins - 1; bb >= 0; --bb) {
    unsigned cb = s_cnt[bb];
    if (c + cb >= krem_in) { bsel = bb; break; }
    c += cb;
    sgt += (double)s_sum[bb];
  }
  *bsel_out = (unsigned)bsel;
  *krem_out = krem_in - c;
  *sgt_out = sgt;
}

// Staging layout inside the arena: [arr(2)][buf(2)][wave][lane] float4
__device__ __forceinline__ int stage_idx(int arr, int buf, int w, int lane) {
  return ((arr * 2 + buf) * NWAVES + w) * 32 + lane;
}

__global__ __launch_bounds__(NTHREADS) void ohem_row_kernel(
    const float* __restrict__ region_label,
    const float* __restrict__ affinity_label,
    const float* __restrict__ region_pred,
    const float* __restrict__ affinity_pred,
    float* __restrict__ row_out)  // [128]: per_batch[64], row_loss_sum[64]
{
  const int r = blockIdx.x;                 // 0..63 : 0-31 region, 32-63 affinity
  const int b = r & 31;
  const float* __restrict__ lab =
      ((r < 32) ? region_label : affinity_label) + (size_t)b * N_PIX;
  const float* __restrict__ prd =
      ((r < 32) ? region_pred : affinity_pred) + (size_t)b * N_PIX;
  const float4* __restrict__ lab4 = (const float4*)lab;
  const float4* __restrict__ prd4 = (const float4*)prd;
  const int NV = N_PIX / 4;                 // 36864

  __shared__ unsigned s_cnt[NBINS];
  __shared__ float    s_sum[NBINS];
  __shared__ __align__(16) char s_arena[ARENA_B];   // staging, later cbuf
  __shared__ double   s_wsp[NWAVES], s_wsn[NWAVES];
  __shared__ unsigned s_wnp[NWAVES];
  __shared__ unsigned sh_np, sh_k, sh_krem, sh_b1, sh_b2, sh_m, sh_cidx;
  __shared__ int      sh_sel;
  __shared__ double   sh_sp, sh_sn, sh_sumgt, sh_topk;

  float4* s_stage = (float4*)s_arena;
  float*  s_cbuf  = (float*)s_arena;

  const int tid  = threadIdx.x;
  const int w    = tid >> 5;                // wave id (wave32)
  const int lane = tid & 31;

  if (tid == 0) sh_cidx = 0u;
  zero_hist(s_cnt, s_sum);
  __syncthreads();

  // ---------------- Pass 1: stats + level-0 histogram (bits 31..21) --------
  // Each wave streams its own contiguous chunk through wave-private LDS
  // staging slots filled by async-to-LDS DMA (double buffered). No barriers
  // needed: a wave consumes only the data it staged itself, and in-order
  // async completion makes s_wait_asynccnt<=2 release the current stage.
  const unsigned long long labU = (unsigned long long)(uintptr_t)lab4;
  const unsigned long long prdU = (unsigned long long)(uintptr_t)prd4;
  const int chunk   = NV / NWAVES;          // 2304 float4 per wave
  const int base    = w * chunk;
  const int NSTAGES = chunk / 32;           // 72

  auto issue_stage = [&](int s, int buf) {
    int i = base + s * 32 + lane;
    unsigned gOff = (unsigned)i * 16u;
    // Low 32 bits of the generic LDS address = offset within the allocation
    // (flat aperture).
    unsigned ldsL = (unsigned)(uintptr_t)&s_stage[stage_idx(0, buf, w, lane)];
    unsigned ldsP = (unsigned)(uintptr_t)&s_stage[stage_idx(1, buf, w, lane)];
    asm volatile("global_load_async_to_lds_b128 %0, %1, %2"
                 :: "v"(ldsL), "v"(gOff), "s"(labU) : "memory");
    asm volatile("global_load_async_to_lds_b128 %0, %1, %2"
                 :: "v"(ldsP), "v"(gOff), "s"(prdU) : "memory");
  };

  double spl = 0.0, snl = 0.0;
  unsigned npl = 0u;

  issue_stage(0, 0);
  for (int s = 0; s < NSTAGES; ++s) {
    int buf = s & 1;
    if (s + 1 < NSTAGES) {
      issue_stage(s + 1, (s + 1) & 1);
      asm volatile("s_wait_asynccnt 0x2" ::: "memory");  // stage s complete
    } else {
      asm volatile("s_wait_asynccnt 0x0" ::: "memory");
    }
    float4 l4 = s_stage[stage_idx(0, buf, w, lane)];
    float4 p4 = s_stage[stage_idx(1, buf, w, lane)];
    float lv[4] = {l4.x, l4.y, l4.z, l4.w};
    float pv[4] = {p4.x, p4.y, p4.z, p4.w};
#pragma unroll
    for (int c = 0; c < 4; ++c) {
      float d = pv[c] - lv[c];
      float x = d * d;                      // loss >= 0 -> bits monotonic
      if (lv[c] >= STANDARD_F) {
        npl++;
        spl += (double)x;
      } else {
        snl += (double)x;
        unsigned u = __float_as_uint(x);
        atomicAdd(&s_cnt[u >> 21], 1u);     // ds_add_u32
        atomicAdd(&s_sum[u >> 21], x);      // ds_add_f32
      }
    }
  }

  // wave32 reduction, then cross-wave via LDS
#pragma unroll
  for (int off = 16; off > 0; off >>= 1) {
    spl += __shfl_down(spl, off);
    snl += __shfl_down(snl, off);
    npl += __shfl_down(npl, off);
  }
  if (lane == 0) { s_wsp[w] = spl; s_wsn[w] = snl; s_wnp[w] = npl; }
  __syncthreads();

  if (tid == 0) {
    double sp = 0.0, sn = 0.0;
    unsigned np = 0u;
    for (int ww = 0; ww < NWAVES; ++ww) { sp += s_wsp[ww]; sn += s_wsn[ww]; np += s_wnp[ww]; }
    sh_sp = sp; sh_sn = sn; sh_np = np;
    unsigned nn = (unsigned)N_PIX - np;
    unsigned k;
    int sel;
    if (np == 0u)                 { k = TOPK_U; sel = 1; }   // top-50 fallback
    else if (nn < RATIO_U * np)   { k = 0u;     sel = 0; }   // mean_neg path
    else                          { k = RATIO_U * np; sel = 1; }  // k <= nn
    sh_k = k; sh_sel = sel;
    if (sel) {
      unsigned bsel, krem; double sgt;
      scan_level(s_cnt, s_sum, NBINS, k, 0.0, &bsel, &krem, &sgt);
      sh_b1 = bsel; sh_krem = krem; sh_sumgt = sgt;
      sh_m = s_cnt[bsel];          // population of threshold bin
    }
  }
  __syncthreads();

  if (sh_sel) {
    const unsigned b1 = sh_b1;
    if (sh_m <= CAP) {
      // ---- Pass 2 (common case): compact threshold-bin elems into LDS ----
      for (int i = tid; i < NV; i += NTHREADS) {
        __builtin_prefetch(&lab4[i + NTHREADS], 0, 0);   // global_prefetch_b8
        __builtin_prefetch(&prd4[i + NTHREADS], 0, 0);
        float4 l4 = lab4[i];
        float4 p4 = prd4[i];
        float lv[4] = {l4.x, l4.y, l4.z, l4.w};
        float pv[4] = {p4.x, p4.y, p4.z, p4.w};
#pragma unroll
        for (int c = 0; c < 4; ++c) {
          if (lv[c] < STANDARD_F) {
            float d = pv[c] - lv[c];
            float x = d * d;
            unsigned u = __float_as_uint(x);
            if ((u >> 21) == b1) {
              unsigned p = atomicAdd(&sh_cidx, 1u);
              s_cbuf[p] = x;
            }
          }
        }
      }
      __syncthreads();
      const int M = (int)sh_cidx;
      // ---- Level 2 from LDS buffer (bits 20..10) ----
      zero_hist(s_cnt, s_sum);
      __syncthreads();
      for (int j = tid; j < M; j += NTHREADS) {
        float x = s_cbuf[j];
        unsigned u = __float_as_uint(x);
        unsigned bin = (u >> 10) & 0x7FFu;
        atomicAdd(&s_cnt[bin], 1u);
        atomicAdd(&s_sum[bin], x);
      }
      __syncthreads();
      if (tid == 0) {
        unsigned bsel, krem; double sgt;
        scan_level(s_cnt, s_sum, NBINS, sh_krem, sh_sumgt, &bsel, &krem, &sgt);
        sh_b2 = bsel; sh_krem = krem; sh_sumgt = sgt;
      }
      __syncthreads();
      // ---- Level 3 from LDS buffer (bits 9..0) ----
      zero_hist(s_cnt, s_sum);
      __syncthreads();
      {
        const unsigned b2v = sh_b2;
        for (int j = tid; j < M; j += NTHREADS) {
          float x = s_cbuf[j];
          unsigned u = __float_as_uint(x);
          if (((u >> 10) & 0x7FFu) == b2v) {
            unsigned bin = u & 0x3FFu;
            atomicAdd(&s_cnt[bin], 1u);
            atomicAdd(&s_sum[bin], x);
          }
        }
      }
      __syncthreads();
      if (tid == 0) {
        unsigned bsel, krem; double sgt;
        scan_level(s_cnt, s_sum, 1024, sh_krem, sh_sumgt, &bsel, &krem, &sgt);
        unsigned uv = (sh_b1 << 21) | (sh_b2 << 10) | bsel;  // exact threshold
        sh_topk = sgt + (double)krem * (double)__uint_as_float(uv);
      }
      __syncthreads();
    } else {
      // ---- Fallback: two more global passes (rare: huge threshold bin) ----
      zero_hist(s_cnt, s_sum);
      __syncthreads();
      for (int i = tid; i < NV; i += NTHREADS) {
        float4 l4 = lab4[i];
        float4 p4 = prd4[i];
        float lv[4] = {l4.x, l4.y, l4.z, l4.w};
        float pv[4] = {p4.x, p4.y, p4.z, p4.w};
#pragma unroll
        for (int c = 0; c < 4; ++c) {
          if (lv[c] < STANDARD_F) {
            float d = pv[c] - lv[c];
            float x = d * d;
            unsigned u = __float_as_uint(x);
            if ((u >> 21) == b1) {
              unsigned bin = (u >> 10) & 0x7FFu;
              atomicAdd(&s_cnt[bin], 1u);
              atomicAdd(&s_sum[bin], x);
            }
          }
        }
      }
      __syncthreads();
      if (tid == 0) {
        unsigned bsel, krem; double sgt;
        scan_level(s_cnt, s_sum, NBINS, sh_krem, sh_sumgt, &bsel, &krem, &sgt);
        sh_b2 = bsel; sh_krem = krem; sh_sumgt = sgt;
      }
      __syncthreads();
      zero_hist(s_cnt, s_sum);
      __syncthreads();
      {
        const unsigned pfx = (sh_b1 << 11) | sh_b2;
        for (int i = tid; i < NV; i += NTHREADS) {
          float4 l4 = lab4[i];
          float4 p4 = prd4[i];
          float lv[4] = {l4.x, l4.y, l4.z, l4.w};
          float pv[4] = {p4.x, p4.y, p4.z, p4.w};
#pragma unroll
          for (int c = 0; c < 4; ++c) {
            if (lv[c] < STANDARD_F) {
              float d = pv[c] - lv[c];
              float x = d * d;
              unsigned u = __float_as_uint(x);
              if ((u >> 10) == pfx) {
                unsigned bin = u & 0x3FFu;
                atomicAdd(&s_cnt[bin], 1u);
                atomicAdd(&s_sum[bin], x);
              }
            }
          }
        }
      }
      __syncthreads();
      if (tid == 0) {
        unsigned bsel, krem; double sgt;
        scan_level(s_cnt, s_sum, 1024, sh_krem, sh_sumgt, &bsel, &krem, &sgt);
        unsigned uv = (sh_b1 << 21) | (sh_b2 << 10) | bsel;
        sh_topk = sgt + (double)krem * (double)__uint_as_float(uv);
      }
      __syncthreads();
    }
  }

  if (tid == 0) {
    double sp = sh_sp, sn = sh_sn;
    unsigned np = sh_np;
    unsigned nn = (unsigned)N_PIX - np;
    double pb;
    if (!sh_sel) {
      pb = sp / (double)(np ? np : 1u) + sn / (double)(nn ? nn : 1u);
    } else if (np == 0u) {
      pb = sh_topk / (double)TOPK_U;                 // global top-50 mean
    } else {
      pb = sp / (double)np + sh_topk / (double)sh_k; // mean_pos + topk_neg_mean
    }
    row_out[r] = (float)pb;
    row_out[64 + r] = (float)(sp + sn);              // row loss sum (for means)
  }
}

__global__ void ohem_final_kernel(const float* __restrict__ row_out,
                                  float* __restrict__ out) {
  if (threadIdx.x == 0) {
    double t = 0.0, sr = 0.0, sa = 0.0;
    for (int r = 0; r < 64; ++r) t += (double)row_out[r];
    for (int r = 0; r < 32; ++r) sr += (double)row_out[64 + r];
    for (int r = 32; r < 64; ++r) sa += (double)row_out[64 + r];
    out[0] = (float)(t / 32.0);                        // region/B + affinity/B
    out[1] = (float)(sr / (32.0 * (double)N_PIX));     // mean(region_loss)
    out[2] = (float)(sa / (32.0 * (double)N_PIX));     // mean(affinity_loss)
  }
}

extern "C" void kernel_launch(void* const* d_in, const int* in_sizes, int n_in,
                              void* d_out, int out_size, void* d_ws, size_t ws_size,
                              hipStream_t stream) {
  (void)in_sizes; (void)n_in; (void)out_size; (void)ws_size;
  const float* region_label   = (const float*)d_in[0];
  const float* affinity_label = (const float*)d_in[1];
  const float* region_pred    = (const float*)d_in[2];
  const float* affinity_pred  = (const float*)d_in[3];
  float* row_out = (float*)d_ws;   // 128 floats, fully overwritten each call
  ohem_row_kernel<<<64, NTHREADS, 0, stream>>>(region_label, affinity_label,
                                               region_pred, affinity_pred, row_out);
  ohem_final_kernel<<<1, 32, 0, stream>>>(row_out, (float*)d_out);
}